// GCN_53266184405568
// MI455X (gfx1250) — compile-verified
//
#include <hip/hip_runtime.h>
#include <hip/hip_bf16.h>

typedef float v2f __attribute__((ext_vector_type(2)));
typedef float v8f __attribute__((ext_vector_type(8)));

#define GCN_N    50000
#define GCN_IN   64
#define GCN_HID  128
#define GCN_OUT  8
#define GCN_G    64

// ---------------- utility fills ----------------
__global__ void fill_kernel(float* __restrict__ p, float v, int n) {
    int i = blockIdx.x * blockDim.x + threadIdx.x;
    if (i < n) p[i] = v;
}

// ---------------- degree / norm ----------------
__global__ void deg_count_kernel(float* __restrict__ deg, const int* __restrict__ dst, int E) {
    int e = blockIdx.x * blockDim.x + threadIdx.x;
    if (e < E) atomicAdd(&deg[dst[e]], 1.0f);
}

__global__ void dinv_kernel(float* __restrict__ deg, int n) {
    int i = blockIdx.x * blockDim.x + threadIdx.x;
    if (i < n) deg[i] = rsqrtf(fmaxf(deg[i], 1.0f));
}

// ---------------- WMMA f32 GEMM: H = X @ W ----------------
// One wave computes RPW row tiles x 1 col tile (16x16 each) with V_WMMA_F32_16X16X4_F32.
// B fragment loaded once per K-step and reused across RPW independent accumulators.
// All strides are compile-time powers of two -> 32-bit shift addressing, no 64-bit muls.
//
// A frag (16x4 f32, 2 VGPRs): lanes 0-15 -> {A[m][k], A[m][k+1]}, lanes 16-31 -> {A[m][k+2], A[m][k+3]}
// B frag (4x16 f32, 2 VGPRs): lanes 0-15 -> {W[k][n], W[k+1][n]}, lanes 16-31 -> {W[k+2][n], W[k+3][n]}
// C/D (16x16 f32, 8 VGPRs):   vgpr v, lanes 0-15 -> D[v][n], lanes 16-31 -> D[8+v][n]
template <int KDIM, int NCOLS, int RPW>
__global__ void gemm_wmma_kernel(const float* __restrict__ X, const float* __restrict__ W,
                                 float* __restrict__ H, int rowGroups) {
    constexpr int colTiles = NCOLS / 16;
    const int wavesPerBlock = blockDim.x >> 5;
    const int waveId = threadIdx.x >> 5;
    const int lane   = threadIdx.x & 31;
    const int gtile  = blockIdx.x * wavesPerBlock + waveId;
    if (gtile >= rowGroups * colTiles) return;   // wave-uniform: EXEC stays all-ones
    const int rg = gtile / colTiles;
    const int ct = gtile - rg * colTiles;
    const int col0 = ct << 4;
    const int lr = lane & 15;                    // row (A) / col (B,C) within tile
    const int hi = lane >> 4;                    // half-wave selector
    const int rowBase = rg * (16 * RPW);

    v8f acc[RPW] = {};

#pragma unroll 4
    for (int k = 0; k < KDIM; k += 4) {
        const float* bp = W + (k + hi * 2) * NCOLS + col0 + lr;
        v2f b; b.x = bp[0]; b.y = bp[NCOLS];
#pragma unroll
        for (int i = 0; i < RPW; ++i) {
            const float* ap = X + (rowBase + i * 16 + lr) * KDIM + (k + hi * 2);
            v2f a = *(const v2f*)ap;             // 8B aligned: row*KDIM, k, hi*2 all even
            acc[i] = __builtin_amdgcn_wmma_f32_16x16x4_f32(
                         /*neg_a=*/false, a, /*neg_b=*/false, b,
                         /*c_mod=*/(short)0, acc[i], /*reuse_a=*/false, /*reuse_b=*/false);
        }
    }

#pragma unroll
    for (int i = 0; i < RPW; ++i) {
        float* outp = H + (rowBase + i * 16 + hi * 8) * NCOLS + col0 + lr;
#pragma unroll
        for (int v = 0; v < 8; ++v) outp[v * NCOLS] = acc[i][v];
    }
}

// ---------------- aggregation ----------------
// acc[n][f] = bias[f] + h[n][f] * dinv[n]^2   (fused bias + self-loop)
__global__ void agg_init_kernel(const float* __restrict__ h, const float* __restrict__ dinv,
                                const float* __restrict__ bias, float* __restrict__ acc,
                                int total) {
    int idx = blockIdx.x * blockDim.x + threadIdx.x;
    if (idx >= total) return;
    int n = idx >> 7;          // HID == 128
    int f = idx & 127;
    float di = dinv[n];
    acc[idx] = bias[f] + h[idx] * di * di;
}

// One wave per edge; lane covers 4 of the 128 features (float4 gather, 4 f32 atomics).
// h and acc are both ~25.6MB -> resident in MI455X's 192MB L2.
__global__ void edge_agg_kernel(const float* __restrict__ h, const float* __restrict__ dinv,
                                const int* __restrict__ src, const int* __restrict__ dst,
                                float* __restrict__ acc, int E) {
    int idx = blockIdx.x * blockDim.x + threadIdx.x;
    int e = idx >> 5;
    if (e >= E) return;
    int lane = idx & 31;
    int s = src[e];
    int d = dst[e];
    float w = dinv[s] * dinv[d];
    const float4* hp = (const float4*)(h + s * GCN_HID) + lane;
    float4 v = *hp;
    float* ap = acc + d * GCN_HID + lane * 4;
    atomicAdd(ap + 0, v.x * w);
    atomicAdd(ap + 1, v.y * w);
    atomicAdd(ap + 2, v.z * w);
    atomicAdd(ap + 3, v.w * w);
}

__global__ void relu_kernel(float* __restrict__ p, int n) {
    int i = blockIdx.x * blockDim.x + threadIdx.x;
    if (i < n) p[i] = fmaxf(p[i], 0.0f);
}

// ---------------- global mean pool ----------------
// batch is sorted: run-length accumulate per thread, flush atomics only on graph change.
// Block handles 64 nodes x 128 features; thread (f, sub) strides 2 nodes at a time.
__global__ void pool_acc_kernel(const float* __restrict__ h, const int* __restrict__ batch,
                                float* __restrict__ pool, int N) {
    int f = threadIdx.x & 127;
    int sub = threadIdx.x >> 7;                // 0..1
    int nodeBase = blockIdx.x * 64;
    float run = 0.0f;
    int curg = -1;
    for (int i = sub; i < 64; i += 2) {
        int n = nodeBase + i;
        if (n >= N) break;
        int g = batch[n];
        if (g != curg) {
            if (curg >= 0) atomicAdd(&pool[curg * GCN_HID + f], run);
            curg = g;
            run = 0.0f;
        }
        run += h[n * GCN_HID + f];
    }
    if (curg >= 0) atomicAdd(&pool[curg * GCN_HID + f], run);
}

__global__ void cnt_acc_kernel(const int* __restrict__ batch, float* __restrict__ cnt, int N) {
    int n = blockIdx.x * blockDim.x + threadIdx.x;
    if (n < N) atomicAdd(&cnt[batch[n]], 1.0f);
}

// out[g][o] = (pool[g]/cnt[g]) . Wh[:,o] + bh[o]
__global__ void final_proj_kernel(const float* __restrict__ pool, const float* __restrict__ cnt,
                                  const float* __restrict__ Wh, const float* __restrict__ bh,
                                  float* __restrict__ out) {
    int t = blockIdx.x * blockDim.x + threadIdx.x;
    if (t >= GCN_G * GCN_OUT) return;
    int g = t >> 3;
    int o = t & 7;
    float inv = 1.0f / fmaxf(cnt[g], 1.0f);
    float s = 0.0f;
    for (int f = 0; f < GCN_HID; ++f)
        s += pool[g * GCN_HID + f] * Wh[f * GCN_OUT + o];
    out[t] = s * inv + bh[o];
}

extern "C" void kernel_launch(void* const* d_in, const int* in_sizes, int n_in,
                              void* d_out, int out_size, void* d_ws, size_t ws_size,
                              hipStream_t stream) {
    const float* x  = (const float*)d_in[0];
    const float* W1 = (const float*)d_in[1];
    const float* b1 = (const float*)d_in[2];
    const float* W2 = (const float*)d_in[3];
    const float* b2 = (const float*)d_in[4];
    const float* W3 = (const float*)d_in[5];
    const float* b3 = (const float*)d_in[6];
    const float* Wh = (const float*)d_in[7];
    const float* bh = (const float*)d_in[8];
    const int*  edge  = (const int*)d_in[9];
    const int*  batch = (const int*)d_in[10];
    float* out = (float*)d_out;

    const int N = GCN_N, HID = GCN_HID;
    const int E = in_sizes[9] / 2;
    const int* src = edge;
    const int* dst = edge + E;

    // workspace layout
    float* bufA = (float*)d_ws;                       // N*HID
    float* bufB = bufA + (size_t)N * HID;             // N*HID
    float* dinv = bufB + (size_t)N * HID;             // N
    float* pool = dinv + N;                           // G*HID
    float* cnt  = pool + (size_t)GCN_G * HID;         // G

    const int BLK = 256;
    constexpr int RPW = 5;                            // 3125 row tiles = 5 * 625
    const int rowGroups = (N / 16) / RPW;             // 625
    const int gemmWaves = rowGroups * (HID / 16);     // 5000 waves
    const int gemmBlocks = (gemmWaves + 3) / 4;       // 4 waves / 128-thread block
    const int nhTotal = N * HID;                      // 6.4M
    const int nhBlocks = (nhTotal + BLK - 1) / BLK;
    const int edgeBlocks = (E * 32 + BLK - 1) / BLK;
    const int poolBlocks = (N + 63) / 64;

    // ---- degrees -> dinv (self-loop contributes 1) ----
    fill_kernel<<<(N + BLK - 1) / BLK, BLK, 0, stream>>>(dinv, 1.0f, N);
    deg_count_kernel<<<(E + BLK - 1) / BLK, BLK, 0, stream>>>(dinv, dst, E);
    dinv_kernel<<<(N + BLK - 1) / BLK, BLK, 0, stream>>>(dinv, N);

    // ---- layer 1: gemm(x,W1) -> bufA; aggregate -> bufB; relu ----
    gemm_wmma_kernel<GCN_IN, GCN_HID, RPW><<<gemmBlocks, 128, 0, stream>>>(x, W1, bufA, rowGroups);
    agg_init_kernel<<<nhBlocks, BLK, 0, stream>>>(bufA, dinv, b1, bufB, nhTotal);
    edge_agg_kernel<<<edgeBlocks, BLK, 0, stream>>>(bufA, dinv, src, dst, bufB, E);
    relu_kernel<<<nhBlocks, BLK, 0, stream>>>(bufB, nhTotal);

    // ---- layer 2: gemm(bufB,W2) -> bufA; aggregate -> bufB; relu ----
    gemm_wmma_kernel<GCN_HID, GCN_HID, RPW><<<gemmBlocks, 128, 0, stream>>>(bufB, W2, bufA, rowGroups);
    agg_init_kernel<<<nhBlocks, BLK, 0, stream>>>(bufA, dinv, b2, bufB, nhTotal);
    edge_agg_kernel<<<edgeBlocks, BLK, 0, stream>>>(bufA, dinv, src, dst, bufB, E);
    relu_kernel<<<nhBlocks, BLK, 0, stream>>>(bufB, nhTotal);

    // ---- layer 3: gemm(bufB,W3) -> bufA; aggregate -> bufB (no relu) ----
    gemm_wmma_kernel<GCN_HID, GCN_HID, RPW><<<gemmBlocks, 128, 0, stream>>>(bufB, W3, bufA, rowGroups);
    agg_init_kernel<<<nhBlocks, BLK, 0, stream>>>(bufA, dinv, b3, bufB, nhTotal);
    edge_agg_kernel<<<edgeBlocks, BLK, 0, stream>>>(bufA, dinv, src, dst, bufB, E);

    // ---- global mean pool + head ----
    fill_kernel<<<(GCN_G * HID + GCN_G + BLK - 1) / BLK, BLK, 0, stream>>>(pool, 0.0f, GCN_G * HID + GCN_G);
    pool_acc_kernel<<<poolBlocks, BLK, 0, stream>>>(bufB, batch, pool, N);
    cnt_acc_kernel<<<(N + BLK - 1) / BLK, BLK, 0, stream>>>(batch, cnt, N);
    final_proj_kernel<<<2, BLK, 0, stream>>>(pool, cnt, Wh, bh, out);
}